// TopKSAE_49503793053987
// MI455X (gfx1250) — compile-verified
//
#include <hip/hip_runtime.h>
#include <hip/hip_bf16.h>

typedef __attribute__((ext_vector_type(2))) float v2f;
typedef __attribute__((ext_vector_type(8))) float v8f;

// Only check for the device builtin during device compilation; the host pass
// (x86) never has amdgcn builtins.
#if defined(__HIP_DEVICE_COMPILE__) && !__has_builtin(__builtin_amdgcn_wmma_f32_16x16x4_f32)
#error "gfx1250 device: __builtin_amdgcn_wmma_f32_16x16x4_f32 not available"
#endif

// ---------------------------------------------------------------------------
// Kernel 1: z = relu((x - b_pre) @ W_enc + b_enc), dense, via V_WMMA_F32_16X16X4_F32
// Block = 256 threads (8 waves, wave32). C tile 128x128, K-chunk 32.
// Wave tiling: 4x2 waves, each wave owns 32x64 (2x4 sixteen-tiles).
// Assumes M%128==0, N%128==0, K%32==0 (8192/32768/1024 here).
// ---------------------------------------------------------------------------
#define BM 128
#define BN 128
#define BK 32
#define LDA 33   // padded LDS stride for A (odd -> conflict-free column reads)
#define LDB 136  // padded LDS stride for B (2*136 % 64 == 16 -> half-waves hit disjoint banks)

__global__ __launch_bounds__(256) void sae_encode_wmma(
    const float* __restrict__ x, const float* __restrict__ b_pre,
    const float* __restrict__ W_enc, const float* __restrict__ b_enc,
    float* __restrict__ z, int M, int N, int K)
{
  __shared__ float As[BM * LDA];   // ~16.9 KB
  __shared__ float Bs[BK * LDB];   // ~17.4 KB

  const int tid  = threadIdx.x;
  const int lane = tid & 31;
  const int wave = tid >> 5;                 // 0..7
  const int m0 = blockIdx.y * BM;
  const int n0 = blockIdx.x * BN;
  const int wm = (wave >> 1) * 32;           // wave row offset in tile
  const int wn = (wave & 1) * 64;            // wave col offset in tile

  const int lm    = lane & 15;               // M/N index within 16
  const int khalf = (lane >> 4) * 2;         // K split: 0 for lanes 0-15, 2 for 16-31

  v8f acc[2][4];
  const v8f zero8 = {0.f,0.f,0.f,0.f,0.f,0.f,0.f,0.f};
#pragma unroll
  for (int i = 0; i < 2; ++i)
#pragma unroll
    for (int j = 0; j < 4; ++j) acc[i][j] = zero8;

  for (int k0 = 0; k0 < K; k0 += BK) {
    // --- stage A tile: (x - b_pre)[m0:m0+128, k0:k0+32] -> As (row-major, padded)
#pragma unroll
    for (int i = 0; i < 4; ++i) {
      int e = tid + i * 256;                 // 1024 float4 tiles total
      int r = e >> 3;                        // row 0..127
      int c = (e & 7) * 4;                   // col 0..28
      float4 xv = *(const float4*)(x + (size_t)(m0 + r) * K + k0 + c);
      float4 bp = *(const float4*)(b_pre + k0 + c);
      As[r * LDA + c + 0] = xv.x - bp.x;
      As[r * LDA + c + 1] = xv.y - bp.y;
      As[r * LDA + c + 2] = xv.z - bp.z;
      As[r * LDA + c + 3] = xv.w - bp.w;
    }
    // --- stage B tile: W_enc[k0:k0+32, n0:n0+128] -> Bs (row-major, padded)
#pragma unroll
    for (int i = 0; i < 4; ++i) {
      int e = tid + i * 256;
      int r = e >> 5;                        // row 0..31
      int c = (e & 31) * 4;                  // col 0..124
      float4 wv = *(const float4*)(W_enc + (size_t)(k0 + r) * N + n0 + c);
      Bs[r * LDB + c + 0] = wv.x;
      Bs[r * LDB + c + 1] = wv.y;
      Bs[r * LDB + c + 2] = wv.z;
      Bs[r * LDB + c + 3] = wv.w;
    }
    // prefetch next-chunk B through GL2 (global_prefetch_b8)
    if (k0 + BK < K) {
      int r = tid >> 5, c = (tid & 31) * 4;
      __builtin_prefetch(W_enc + (size_t)(k0 + BK + r) * N + n0 + c, 0, 1);
    }
    __syncthreads();

    // --- compute: 8 K-steps of 4, 2x4 sixteen-tiles per wave -> 64 WMMAs/chunk/wave
#pragma unroll
    for (int kk = 0; kk < BK; kk += 4) {
      v2f a[2], b[4];
#pragma unroll
      for (int i = 0; i < 2; ++i) {
        int row = wm + i * 16 + lm;
        a[i].x = As[row * LDA + kk + khalf];
        a[i].y = As[row * LDA + kk + khalf + 1];
      }
#pragma unroll
      for (int j = 0; j < 4; ++j) {
        int col = wn + j * 16 + lm;
        b[j].x = Bs[(kk + khalf)     * LDB + col];
        b[j].y = Bs[(kk + khalf + 1) * LDB + col];
      }
#if defined(__HIP_DEVICE_COMPILE__)
#pragma unroll
      for (int i = 0; i < 2; ++i)
#pragma unroll
        for (int j = 0; j < 4; ++j)
          acc[i][j] = __builtin_amdgcn_wmma_f32_16x16x4_f32(
              false, a[i], false, b[j], (short)0, acc[i][j], false, false);
#endif
    }
    __syncthreads();
  }

  // --- epilogue: + b_enc, ReLU, store. C layout: VGPR r holds M=r (lanes 0-15)
  // and M=r+8 (lanes 16-31); N = lane&15.
#pragma unroll
  for (int i = 0; i < 2; ++i)
#pragma unroll
    for (int j = 0; j < 4; ++j) {
      int col = n0 + wn + j * 16 + (lane & 15);
      float be = b_enc[col];
#pragma unroll
      for (int r = 0; r < 8; ++r) {
        int row = m0 + wm + i * 16 + r + ((lane >> 4) * 8);
        float v = acc[i][j][r] + be;
        z[(size_t)row * N + col] = fmaxf(v, 0.0f);
      }
    }
}

// ---------------------------------------------------------------------------
// Kernel 2: per-row exact top-k (radix-256 select on float bit patterns;
// ReLU => non-negative => uint order == float order), index-ordered tie break,
// in-place sparsification of z, and fused sparse decode:
//   xhat[row,:] = sum_j v_j * W_dec[idx_j,:] + b_pre
// One 256-thread workgroup per row; row (128 KB) staged in dynamic LDS
// (CDNA5: up to 320 KB per workgroup).
// ---------------------------------------------------------------------------
#define K2_THREADS 256
#define MAXK 128

__global__ __launch_bounds__(K2_THREADS) void sae_topk_decode(
    const float* __restrict__ b_pre, const float* __restrict__ W_dec,
    float* __restrict__ z, float* __restrict__ xhat,
    int N, int Dout, const int* __restrict__ kptr)
{
  extern __shared__ unsigned char smem_raw[];
  float*    rowf = (float*)smem_raw;       // N floats
  unsigned* rowu = (unsigned*)smem_raw;

  __shared__ unsigned hist[256];
  __shared__ unsigned scanT[K2_THREADS + 1];
  __shared__ unsigned scanP[K2_THREADS + 1];
  __shared__ unsigned sel_digit, sel_rem;
  __shared__ int   pair_idx[MAXK];
  __shared__ float pair_val[MAXK];

  const int tid = threadIdx.x;
  const int row = blockIdx.x;
  int k = *kptr; if (k > MAXK) k = MAXK;
  const size_t rowoff = (size_t)row * N;
  const int chunk = N / K2_THREADS;        // 128
  const int base  = tid * chunk;

  // stage dense z row into LDS (coalesced float4)
  {
    const float4* zin4 = (const float4*)(z + rowoff);
    float4* row4 = (float4*)rowf;
    for (int i = tid; i < N / 4; i += K2_THREADS) row4[i] = zin4[i];
  }
  __syncthreads();

  // ---- 4x 8-bit MSD radix select for the k-th largest value
  unsigned prefix = 0, remaining = (unsigned)k;
  for (int shift = 24; shift >= 0; shift -= 8) {
    hist[tid] = 0;
    __syncthreads();
    unsigned hm = (shift == 24) ? 0u : (0xFFFFFFFFu << (shift + 8));
    for (int i = 0; i < chunk; ++i) {
      unsigned u = rowu[base + i];
      if ((u & hm) == (prefix & hm))
        atomicAdd(&hist[(u >> shift) & 255u], 1u);
    }
    __syncthreads();
    if (tid == 0) {
      unsigned cum = 0; int b = 255;
      for (; b > 0; --b) {
        if (cum + hist[b] >= remaining) break;
        cum += hist[b];
      }
      sel_digit = (unsigned)b;
      sel_rem   = remaining - cum;
    }
    __syncthreads();
    prefix |= sel_digit << shift;
    remaining = sel_rem;
    __syncthreads();
  }
  const unsigned thr = prefix;   // bit pattern of k-th largest; `remaining` = #ties to keep

  // ---- per-thread counts over contiguous chunk (global index order)
  unsigned tcnt = 0, gcnt = 0;
  for (int i = 0; i < chunk; ++i) {
    unsigned u = rowu[base + i];
    gcnt += (u > thr);
    tcnt += (u == thr);
  }
  scanT[tid] = tcnt;
  __syncthreads();
  if (tid == 0) {
    unsigned s = 0;
    for (int t = 0; t < K2_THREADS; ++t) { unsigned c = scanT[t]; scanT[t] = s; s += c; }
    scanT[K2_THREADS] = s;
  }
  __syncthreads();
  const unsigned tbase = scanT[tid];

  unsigned my_ties = 0;
  if (tbase < remaining) {
    unsigned room = remaining - tbase;
    my_ties = (room < tcnt) ? room : tcnt;
  }
  // pairs appended for decode (strictly positive values only)
  unsigned pos_keep = gcnt + ((thr > 0u) ? my_ties : 0u);
  scanP[tid] = pos_keep;
  __syncthreads();
  if (tid == 0) {
    unsigned s = 0;
    for (int t = 0; t < K2_THREADS; ++t) { unsigned c = scanP[t]; scanP[t] = s; s += c; }
    scanP[K2_THREADS] = s;
  }
  __syncthreads();
  unsigned pslot = scanP[tid];
  const unsigned nkept = scanP[K2_THREADS];

  // ---- rewrite row sparse in place + collect (idx, val) pairs deterministically
  unsigned tie_seen = 0;
  for (int i = 0; i < chunk; ++i) {
    int e = base + i;
    unsigned u = rowu[e];
    bool keep;
    if (u > thr) keep = true;
    else if (u == thr) { keep = (tbase + tie_seen) < remaining; tie_seen++; }
    else keep = false;
    float v = keep ? rowf[e] : 0.0f;
    z[rowoff + e] = v;
    if (keep && u > 0u && pslot < MAXK) {
      pair_idx[pslot] = e;
      pair_val[pslot] = rowf[e];
      pslot++;
    }
  }
  __syncthreads();

  // ---- sparse decode: each thread owns 4 output columns (Dout == 1024)
  {
    const int c0 = tid * 4;
    float4 acc = *(const float4*)(b_pre + c0);
    int nk = (int)nkept; if (nk > MAXK) nk = MAXK;
    for (int j = 0; j < nk; ++j) {
      float v = pair_val[j];
      const float4 w = *(const float4*)(W_dec + (size_t)pair_idx[j] * Dout + c0);
      acc.x += v * w.x; acc.y += v * w.y; acc.z += v * w.z; acc.w += v * w.w;
    }
    *(float4*)(xhat + (size_t)row * Dout + c0) = acc;
  }
}

// ---------------------------------------------------------------------------
// Launch: inputs = x, b_pre, W_enc, b_enc, W_dec, k
// d_out = [x_hat (B x D_IN) | z_sparse (B x D_SAE)] concatenated flat.
// Kernel 1 writes dense z into the z_sparse region (scratch-in-place);
// kernel 2 stages each row to LDS, then finalizes it sparse + writes x_hat.
// ---------------------------------------------------------------------------
extern "C" void kernel_launch(void* const* d_in, const int* in_sizes, int n_in,
                              void* d_out, int out_size, void* d_ws, size_t ws_size,
                              hipStream_t stream) {
  const float* x     = (const float*)d_in[0];
  const float* b_pre = (const float*)d_in[1];
  const float* W_enc = (const float*)d_in[2];
  const float* b_enc = (const float*)d_in[3];
  const float* W_dec = (const float*)d_in[4];
  const int*   kptr  = (const int*)d_in[5];

  const int D_IN  = in_sizes[1];             // 1024
  const int D_SAE = in_sizes[3];             // 32768
  const int B     = in_sizes[0] / D_IN;      // 8192

  float* out  = (float*)d_out;
  float* xhat = out;
  float* z    = out + (size_t)B * D_IN;

  dim3 g1(D_SAE / BN, B / BM);
  sae_encode_wmma<<<g1, 256, 0, stream>>>(x, b_pre, W_enc, b_enc, z, B, D_SAE, D_IN);

  size_t shmem = (size_t)D_SAE * sizeof(float);  // 128 KB row buffer
  sae_topk_decode<<<B, K2_THREADS, shmem, stream>>>(b_pre, W_dec, z, xhat,
                                                    D_SAE, D_IN, kptr);
}